// RPN_17523466568010
// MI455X (gfx1250) — compile-verified
//
#include <hip/hip_runtime.h>
#include <hip/hip_bf16.h>
#include <math.h>

typedef float v2f __attribute__((ext_vector_type(2)));
typedef float v8f __attribute__((ext_vector_type(8)));

#define FH 160
#define FW 160
#define CIN 512
#define IHW 2560.0f
#define NANCH (FH*FW*9)          // 230400
#define NPIX (FH*FW)             // 25600
#define NHEAD 144                // 72 reg + 9 cls + 54 lm + 9 pad

// ---- ws layout (floats) ----
#define X_OFF      0ull                           // 25600*512 = 13107200
#define W2_OFF     13107200ull                    // 512*144   = 73728
#define B2_OFF     (W2_OFF + 73728ull)            // 144
#define DELT_OFF   (B2_OFF + 144ull)              // 230400*4  = 921600
#define SCORE_OFF  (DELT_OFF + 921600ull)         // 230400
// after heads GEMM, x region is dead -> overlay:
#define BOX_OFF    0ull                           // 230400*4
#define ALIVE_OFF  921600ull                      // 230400

// ---- d_out layout (floats) ----
#define OUT_ROIS   0
#define OUT_LM     24
#define OUT_ANCH   (24 + NPIX*54)                 // 1382424

// ---------------------------------------------------------------- anchors
__global__ __launch_bounds__(256) void anchors_kernel(float* __restrict__ out_anchors) {
    int n = blockIdx.x * 256 + threadIdx.x;
    if (n >= NANCH) return;
    const float RAT[3] = {0.5f, 1.0f, 2.0f};
    const float SCL[3] = {8.0f, 16.0f, 32.0f};
    int p = n / 9, a = n % 9;
    int i = p / FW, j = p % FW;
    float r = RAT[a / 3], s = SCL[a % 3];
    float h = sqrtf(s * s / r) * 16.0f;
    float w = h * r;                   // x_stride/y_stride == 1
    float cx = (j + 0.5f) * 16.0f;
    float cy = (i + 0.5f) * 16.0f;
    float* o = out_anchors + (size_t)n * 4;
    o[0] = cx - w * 0.5f;
    o[1] = cy - h * 0.5f;
    o[2] = cx + w * 0.5f;
    o[3] = cy + h * 0.5f;
}

// ---------------------------------------------------------------- pack head weights
__global__ __launch_bounds__(256) void pack_kernel(const float* __restrict__ reg_w,
                                                   const float* __restrict__ cls_w,
                                                   const float* __restrict__ lm_w,
                                                   const float* __restrict__ reg_b,
                                                   const float* __restrict__ cls_b,
                                                   const float* __restrict__ lm_b,
                                                   float* __restrict__ W2,
                                                   float* __restrict__ B2) {
    int idx = blockIdx.x * 256 + threadIdx.x;
    if (idx < CIN * NHEAD) {
        int k = idx / NHEAD, c = idx % NHEAD;
        float v = 0.0f;
        if (c < 72)       v = reg_w[k * 72 + c];
        else if (c < 81)  v = cls_w[k * 9 + (c - 72)];
        else if (c < 135) v = lm_w[k * 54 + (c - 81)];
        W2[idx] = v;
    }
    if (idx < NHEAD) {
        float v = 0.0f;
        if (idx < 72)       v = reg_b[idx];
        else if (idx < 81)  v = cls_b[idx - 72];
        else if (idx < 135) v = lm_b[idx - 81];
        B2[idx] = v;
    }
}

// ---------------------------------------------------------------- conv 3x3 (implicit GEMM, WMMA f32)
// grid: (160*5, 4), block 256 (8 waves). Tile: M=32 pixels (strip along W), N=128 co.
__global__ __launch_bounds__(256) void conv3x3_wmma(const float* __restrict__ F,
                                                    const float* __restrict__ W1,
                                                    const float* __restrict__ B1,
                                                    float* __restrict__ X) {
    __shared__ float As[32 * 68];     // pixels x ci-chunk, padded stride 68
    __shared__ float Bs[64 * 144];    // ci x co, stride 144 (bank friendly)

    const int tid  = threadIdx.x;
    const int lane = tid & 31, wave = tid >> 5;
    const int waveM = wave & 1, waveN = wave >> 1;   // waveN 0..3
    const int lh = lane >> 4, lm = lane & 15;

    const int h  = blockIdx.x / 5;
    const int w0 = (blockIdx.x % 5) * 32;
    const int co0 = blockIdx.y * 128;

    v8f acc0 = {}, acc1 = {};

    for (int kh = 0; kh < 3; ++kh) {
        const int row = h + kh - 1;
        const bool rowOK = (row >= 0) && (row < FH);
        for (int kw = 0; kw < 3; ++kw) {
            for (int cc = 0; cc < CIN; cc += 64) {
                // stage A: 32 pixels x 64 ci  (512 float4)
                for (int e = tid; e < 512; e += 256) {
                    int m  = e >> 4;
                    int c4 = (e & 15) << 2;
                    int col = w0 + m + kw - 1;
                    float4 v = make_float4(0.f, 0.f, 0.f, 0.f);
                    if (rowOK && col >= 0 && col < FW)
                        v = *(const float4*)&F[((size_t)(row * FW + col) * CIN) + cc + c4];
                    *(float4*)&As[m * 68 + c4] = v;
                }
                // stage B: 64 ci x 128 co  (2048 float4)
                const float* wsrc = &W1[((size_t)((kh * 3 + kw) * CIN + cc)) * CIN + co0];
                for (int e = tid; e < 2048; e += 256) {
                    int k  = e >> 5;
                    int c4 = (e & 31) << 2;
                    *(float4*)&Bs[k * 144 + c4] = *(const float4*)&wsrc[(size_t)k * CIN + c4];
                }
                __syncthreads();

                const int am = (waveM * 16 + lm) * 68;
                const int bn = waveN * 32 + lm;
                #pragma unroll
                for (int k = 0; k < 64; k += 4) {
                    v2f a = *(const v2f*)&As[am + k + lh * 2];
                    v2f b0, b1;
                    int r0 = (k + lh * 2) * 144, r1 = r0 + 144;
                    b0.x = Bs[r0 + bn];      b0.y = Bs[r1 + bn];
                    b1.x = Bs[r0 + bn + 16]; b1.y = Bs[r1 + bn + 16];
                    acc0 = __builtin_amdgcn_wmma_f32_16x16x4_f32(false, a, false, b0,
                                                                 (short)0, acc0, false, false);
                    acc1 = __builtin_amdgcn_wmma_f32_16x16x4_f32(false, a, false, b1,
                                                                 (short)0, acc1, false, false);
                }
                __syncthreads();
            }
        }
    }

    // epilogue: bias + relu, store x[p][co]
    #pragma unroll
    for (int r = 0; r < 8; ++r) {
        int m_l = r + lh * 8;
        int p   = h * FW + (w0 + waveM * 16 + m_l);
        int coA = co0 + waveN * 32 + lm;
        float va = acc0[r] + B1[coA];
        float vb = acc1[r] + B1[coA + 16];
        X[(size_t)p * CIN + coA]      = va > 0.f ? va : 0.f;
        X[(size_t)p * CIN + coA + 16] = vb > 0.f ? vb : 0.f;
    }
}

// ---------------------------------------------------------------- heads GEMM (x @ W2), WMMA f32
// grid: 400, block 128 (4 waves). Tile: M=64 pixels, N=144.
__global__ __launch_bounds__(128) void heads_wmma(const float* __restrict__ X,
                                                  const float* __restrict__ W2,
                                                  const float* __restrict__ B2,
                                                  float* __restrict__ deltas,
                                                  float* __restrict__ scores,
                                                  float* __restrict__ lm_out) {
    __shared__ float As[64 * 68];
    __shared__ float Bs[64 * 144];

    const int tid  = threadIdx.x;
    const int lane = tid & 31, wave = tid >> 5;
    const int lh = lane >> 4, lm = lane & 15;
    const int p0 = blockIdx.x * 64;

    v8f acc[9];
    #pragma unroll
    for (int j = 0; j < 9; ++j) acc[j] = (v8f){};

    for (int cc = 0; cc < CIN; cc += 64) {
        for (int e = tid; e < 1024; e += 128) {     // 64x64 A
            int m = e >> 4, c4 = (e & 15) << 2;
            *(float4*)&As[m * 68 + c4] =
                *(const float4*)&X[(size_t)(p0 + m) * CIN + cc + c4];
        }
        for (int e = tid; e < 2304; e += 128) {     // 64x144 B
            int k = e / 36, c4 = (e % 36) << 2;
            *(float4*)&Bs[k * 144 + c4] = *(const float4*)&W2[(size_t)(cc + k) * NHEAD + c4];
        }
        __syncthreads();

        const int am = (wave * 16 + lm) * 68;
        for (int k = 0; k < 64; k += 4) {
            v2f a = *(const v2f*)&As[am + k + lh * 2];
            int r0 = (k + lh * 2) * 144, r1 = r0 + 144;
            #pragma unroll
            for (int j = 0; j < 9; ++j) {
                v2f b;
                b.x = Bs[r0 + j * 16 + lm];
                b.y = Bs[r1 + j * 16 + lm];
                acc[j] = __builtin_amdgcn_wmma_f32_16x16x4_f32(false, a, false, b,
                                                               (short)0, acc[j], false, false);
            }
        }
        __syncthreads();
    }

    // epilogue: route channels
    #pragma unroll
    for (int j = 0; j < 9; ++j) {
        int co = j * 16 + lm;
        float bias = B2[co];
        #pragma unroll
        for (int r = 0; r < 8; ++r) {
            int m_l = r + lh * 8;
            int p   = p0 + wave * 16 + m_l;
            float val = acc[j][r] + bias;
            if (co < 72) {
                int a_ = co >> 3, c = co & 7;
                if (c < 4) deltas[((size_t)p * 9 + a_) * 4 + c] = val;
            } else if (co < 81) {
                scores[(size_t)p * 9 + (co - 72)] = 1.0f / (1.0f + expf(-val));
            } else if (co < 135) {
                lm_out[(size_t)p * 54 + (co - 81)] = val;
            }
        }
    }
}

// ---------------------------------------------------------------- decode boxes + validity
__global__ __launch_bounds__(256) void decode_kernel(const float* __restrict__ anchors,
                                                     const float* __restrict__ deltas,
                                                     float* __restrict__ boxes,
                                                     float* __restrict__ alive) {
    int n = blockIdx.x * 256 + threadIdx.x;
    if (n >= NANCH) return;
    const float* a = anchors + (size_t)n * 4;
    float ax1 = a[0], ay1 = a[1], ax2 = a[2], ay2 = a[3];
    float wa = ax2 - ax1, ha = ay2 - ay1;
    float xa = ax1 + wa * 0.5f, ya = ay1 + ha * 0.5f;
    const float* d = deltas + (size_t)n * 4;
    float px = xa + wa * d[0];
    float py = ya + ha * d[1];
    float pw = wa * expf(d[2]);
    float ph = ha * expf(d[3]);
    float x1 = px - pw * 0.5f, y1 = py - ph * 0.5f;
    float* b = boxes + (size_t)n * 4;
    b[0] = fminf(fmaxf(x1, 0.f), IHW);
    b[1] = fminf(fmaxf(y1, 0.f), IHW);
    b[2] = fminf(fmaxf(x1 + pw, 0.f), IHW);
    b[3] = fminf(fmaxf(y1 + ph, 0.f), IHW);
    bool valid = (ax1 >= 0.f) && (ay1 >= 0.f) && (ax2 <= IHW) && (ay2 <= IHW);
    alive[n] = valid ? 1.0f : 0.0f;
}

// ---------------------------------------------------------------- greedy NMS (single workgroup)
__global__ __launch_bounds__(1024) void nms_kernel(const float* __restrict__ boxes,
                                                   const float* __restrict__ scores,
                                                   float* __restrict__ alive,
                                                   float* __restrict__ rois) {
    __shared__ float s_v[1024];
    __shared__ int   s_i[1024];
    __shared__ float s_box[4];
    const int tid = threadIdx.x;

    for (int it = 0; it < 6; ++it) {
        // local argmax over alive entries (first-index tie-break, fallback idx 0)
        float bv = -1.0f; int bi = 0;
        for (int n = tid; n < NANCH; n += 1024) {
            if (alive[n] > 0.5f) {
                float v = scores[n];
                if (v > bv || (v == bv && n < bi)) { bv = v; bi = n; }
            }
        }
        s_v[tid] = bv; s_i[tid] = bi;
        __syncthreads();
        for (int off = 512; off > 0; off >>= 1) {
            if (tid < off) {
                float v2 = s_v[tid + off]; int i2 = s_i[tid + off];
                if (v2 > s_v[tid] || (v2 == s_v[tid] && i2 < s_i[tid])) {
                    s_v[tid] = v2; s_i[tid] = i2;
                }
            }
            __syncthreads();
        }
        int sel = s_i[0];
        if (tid == 0) {
            #pragma unroll
            for (int c = 0; c < 4; ++c) {
                float bc = boxes[(size_t)sel * 4 + c];
                s_box[c] = bc;
                rois[it * 4 + c] = bc;
            }
        }
        __syncthreads();
        float bx1 = s_box[0], by1 = s_box[1], bx2 = s_box[2], by2 = s_box[3];
        float areaB = (bx2 - bx1) * (by2 - by1);
        for (int n = tid; n < NANCH; n += 1024) {
            if (alive[n] > 0.5f) {
                const float* q = boxes + (size_t)n * 4;
                float ix1 = fmaxf(bx1, q[0]);
                float iy1 = fmaxf(by1, q[1]);
                float ix2 = fminf(bx2, q[2]);
                float iy2 = fminf(by2, q[3]);
                float inter = fmaxf(ix2 - ix1, 0.f) * fmaxf(iy2 - iy1, 0.f);
                float areaQ = (q[2] - q[0]) * (q[3] - q[1]);
                float iou = inter / (areaB + areaQ - inter + 1e-9f);
                if (iou > 0.5f) alive[n] = 0.0f;
            }
        }
        __syncthreads();
    }
}

// ---------------------------------------------------------------- launcher
extern "C" void kernel_launch(void* const* d_in, const int* in_sizes, int n_in,
                              void* d_out, int out_size, void* d_ws, size_t ws_size,
                              hipStream_t stream) {
    (void)in_sizes; (void)n_in; (void)out_size; (void)ws_size;
    const float* features = (const float*)d_in[0];
    // d_in[1] = image (shape only, unused)
    const float* conv1_w  = (const float*)d_in[2];
    const float* conv1_b  = (const float*)d_in[3];
    const float* reg_w    = (const float*)d_in[4];
    const float* reg_b    = (const float*)d_in[5];
    const float* cls_w    = (const float*)d_in[6];
    const float* cls_b    = (const float*)d_in[7];
    const float* lm_w     = (const float*)d_in[8];
    const float* lm_b     = (const float*)d_in[9];

    float* out = (float*)d_out;
    float* ws  = (float*)d_ws;

    float* X      = ws + X_OFF;
    float* W2     = ws + W2_OFF;
    float* B2     = ws + B2_OFF;
    float* deltas = ws + DELT_OFF;
    float* scores = ws + SCORE_OFF;
    float* boxes  = ws + BOX_OFF;     // overlays X after heads GEMM
    float* alive  = ws + ALIVE_OFF;   // overlays X after heads GEMM

    float* out_rois = out + OUT_ROIS;
    float* out_lm   = out + OUT_LM;
    float* out_anch = out + OUT_ANCH;

    anchors_kernel<<<NANCH / 256, 256, 0, stream>>>(out_anch);
    pack_kernel<<<(CIN * NHEAD + 255) / 256, 256, 0, stream>>>(
        reg_w, cls_w, lm_w, reg_b, cls_b, lm_b, W2, B2);
    conv3x3_wmma<<<dim3(FH * 5, 4), 256, 0, stream>>>(features, conv1_w, conv1_b, X);
    heads_wmma<<<NPIX / 64, 128, 0, stream>>>(X, W2, B2, deltas, scores, out_lm);
    decode_kernel<<<NANCH / 256, 256, 0, stream>>>(out_anch, deltas, boxes, alive);
    nms_kernel<<<1, 1024, 0, stream>>>(boxes, scores, alive, out_rois);
}